// diff_pool_net1_3728031613303
// MI455X (gfx1250) — compile-verified
//
#include <hip/hip_runtime.h>

// ---------------------------------------------------------------------------
// DiffPool GCN for MI455X (gfx1250, wave32, WMMA).
// All dense GEMMs use V_WMMA_F32_16X16X4_F32 (exact fp32 semantics).
// adj (256MB) is streamed exactly once: t = adj@s fused with sum(adj^2),
// using a padded/transposed s so all fragment loads are contiguous b64 with
// no per-element guards (clamped addresses; garbage rows/cols never stored).
// link-loss uses sum(adj^2) - 2*trace(s'adj s) + ||s's||_F^2 (no NxN temp).
// ---------------------------------------------------------------------------

typedef float v2f __attribute__((ext_vector_type(2)));
typedef float v8f __attribute__((ext_vector_type(8)));

static constexpr int N_    = 8192;
static constexpr int F_IN  = 512;
static constexpr int E_    = 262144;
static constexpr int C_    = 100;
static constexpr int CPAD_ = 112;   // 7 tiles of 16
static constexpr int NH_   = 16;
static constexpr int NO0_  = 10;
static constexpr int NCLS_ = 10;

__device__ __forceinline__ v8f wmma_f32_16x16x4(v2f a, v2f b, v8f c) {
#if defined(__AMDGCN__) && __has_builtin(__builtin_amdgcn_wmma_f32_16x16x4_f32)
  // 8 args: (neg_a, A, neg_b, B, c_mod, C, reuse_a, reuse_b)
  return __builtin_amdgcn_wmma_f32_16x16x4_f32(false, a, false, b, (short)0, c,
                                               false, false);
#else
  c[0] += a.x * b.x + a.y * b.y;   // host-pass / fallback placeholder
  return c;
#endif
}

// ---------------------------------------------------------------------------
// Generic WMMA GEMM: C = A @ B (+bias, +relu).  One 16x16 C-tile per wave.
// A: MxK row-major.  B: KxN row-major.  C: MxN row-major.
// MODE: 0 = store, 1 = atomicAdd (k-split via gridDim.y), 2 = read-add-store.
// Out-of-range M rows / N cols are address-clamped (results discarded at the
// guarded writeback), so the main loop has NO conditional loads. Only a K%4
// tail (K=10 case) needs zero-masked elements.
// Fragment layouts per CDNA5 ISA 7.12.2 (f32):
//   A 16x4 : lane l -> m=l%16, VGPRj holds K = j + 2*(l/16)
//   B 4x16 : lane l -> n=l%16, VGPRj holds K = j + 2*(l/16)
//   C 16x16: VGPR r -> row r + 8*(l/16), col l%16
// ---------------------------------------------------------------------------
template <int MODE, bool BIAS, bool RELU>
__global__ __launch_bounds__(128) void gemm_wmma(
    const float* __restrict__ A, const float* __restrict__ B,
    const float* __restrict__ bias, float* __restrict__ C,
    int M, int N, int K, int lda, int ldb, int ldc) {
  const int lane = threadIdx.x & 31;
  const int wave = blockIdx.x * (blockDim.x >> 5) + (threadIdx.x >> 5);
  const int tilesN = (N + 15) >> 4;
  const int tilesM = (M + 15) >> 4;
  if (wave >= tilesM * tilesN) return;
  const int tm = wave / tilesN, tn = wave % tilesN;
  const int m0 = tm << 4, n0 = tn << 4;
  const int r = lane & 15, hi = lane >> 4;

  // k-split bounds (gridDim.y chunks, each a multiple of 4)
  const int Kc = (K + 3) & ~3;
  const int chunk = (((Kc >> 2) + gridDim.y - 1) / gridDim.y) << 2;
  const int kb = blockIdx.y * chunk;
  const int ke = min(kb + chunk, Kc);
  const int kAligned = K & ~3;
  const int mainEnd = min(ke, kAligned);

  const int amc = min(m0 + r, M - 1);  // clamped: OOB rows never stored
  const int bnc = min(n0 + r, N - 1);  // clamped: OOB cols never stored

  v8f acc = {};
  // contiguous A fragment (lda even for all call sites -> 8B aligned)
  const float* ap = A + (long long)amc * lda + kb + 2 * hi;
  const float* bp = B + (long long)(kb + 2 * hi) * ldb + bnc;
  for (int k = kb; k < mainEnd; k += 4) {
    v2f a = *(const v2f*)ap;
    v2f b;
    b.x = bp[0];
    b.y = bp[ldb];
    acc = wmma_f32_16x16x4(a, b, acc);
    ap += 4;
    bp += (long long)4 * ldb;
  }
  // K tail (only when K%4 != 0 and this chunk owns it)
  if ((K & 3) && kAligned >= kb && kAligned < ke) {
    const int ka = kAligned + 2 * hi;
    const int k0 = min(ka, K - 1), k1 = min(ka + 1, K - 1);
    float a0 = A[(long long)amc * lda + k0];
    float a1 = A[(long long)amc * lda + k1];
    float b0 = B[(long long)k0 * ldb + bnc];
    float b1 = B[(long long)k1 * ldb + bnc];
    v2f a, b;
    a.x = (ka < K) ? a0 : 0.f;
    a.y = (ka + 1 < K) ? a1 : 0.f;
    b.x = (ka < K) ? b0 : 0.f;
    b.y = (ka + 1 < K) ? b1 : 0.f;
    acc = wmma_f32_16x16x4(a, b, acc);
  }

#pragma unroll
  for (int rr = 0; rr < 8; ++rr) {
    const int row = m0 + rr + 8 * hi;
    const int col = n0 + r;
    if (row < M && col < N) {
      float v = acc[rr];
      if (BIAS) v += bias[col];
      if (RELU) v = fmaxf(v, 0.f);
      const long long off = (long long)row * ldc + col;
      if (MODE == 0)      C[off] = v;
      else if (MODE == 1) atomicAdd(&C[off], v);
      else                C[off] += v;
    }
  }
}

template <int MODE, bool BIAS, bool RELU>
static void run_gemm(const float* A, const float* B, const float* bias, float* C,
                     int M, int N, int K, int lda, int ldb, int ldc,
                     int kChunks, hipStream_t stream) {
  const int tiles = ((M + 15) / 16) * ((N + 15) / 16);
  dim3 grid((tiles + 3) / 4, kChunks);
  gemm_wmma<MODE, BIAS, RELU>
      <<<grid, 128, 0, stream>>>(A, B, bias, C, M, N, K, lda, ldb, ldc);
}

// ---------------------------------------------------------------------------
// Fused t = adj @ s  (+ sum(adj^2)).  One wave = 16 adj rows x all 7 N-tiles.
// B comes from sTp (padded, transposed s: [112][8192]) so every fragment is a
// contiguous b64 load; the 7 tiles are immediate offsets off one base pointer.
// k-split so each adj element is loaded exactly once across the grid.
// ---------------------------------------------------------------------------
__global__ __launch_bounds__(128) void adj_s_wmma(
    const float* __restrict__ adj, const float* __restrict__ sTp,
    float* __restrict__ t, float* __restrict__ sum_adj2, int kChunkLen) {
  const int lane = threadIdx.x & 31;
  const int wave = blockIdx.x * (blockDim.x >> 5) + (threadIdx.x >> 5);
  const int tilesM = N_ >> 4;                 // 512
  const int tm = wave % tilesM;
  const int kb = (wave / tilesM) * kChunkLen;
  const int r = lane & 15, hi = lane >> 4;

  v8f acc[7] = {};
  float sq = 0.f;
  const float* ap = adj + (long long)(tm * 16 + r) * N_ + kb + 2 * hi;
  const float* bp = sTp + (long long)r * N_ + kb + 2 * hi;
#pragma unroll 2
  for (int k = 0; k < kChunkLen; k += 4) {
    v2f a = *(const v2f*)ap;
    sq += a.x * a.x + a.y * a.y;
#pragma unroll
    for (int tn = 0; tn < 7; ++tn) {
      v2f b = *(const v2f*)(bp + (long long)tn * 16 * N_);
      acc[tn] = wmma_f32_16x16x4(a, b, acc[tn]);
    }
    ap += 4;
    bp += 4;
  }
#pragma unroll
  for (int tn = 0; tn < 7; ++tn)
#pragma unroll
    for (int rr = 0; rr < 8; ++rr) {
      const int row = tm * 16 + rr + 8 * hi;
      const int col = tn * 16 + r;
      if (col < C_) atomicAdd(&t[(long long)row * C_ + col], acc[tn][rr]);
    }
  for (int o = 16; o; o >>= 1) sq += __shfl_xor(sq, o, 32);
  if (lane == 0) atomicAdd(sum_adj2, sq);
}

// sTp[n][k] = (n < cols) ? s[k][n] : 0   — padded transpose, [CPAD_][N_]
__global__ void transpose_pad_kernel(const float* __restrict__ s,
                                     float* __restrict__ sTp) {
  long long total = (long long)CPAD_ * N_;
  for (long long idx = (long long)blockIdx.x * blockDim.x + threadIdx.x;
       idx < total; idx += (long long)gridDim.x * blockDim.x) {
    int nn = (int)(idx / N_);
    int k = (int)(idx % N_);
    sTp[idx] = (nn < C_) ? s[(long long)k * C_ + nn] : 0.f;
  }
}

// ------------------------------- utility kernels ---------------------------
__global__ void fill_kernel(float* __restrict__ p, float v, long long n) {
  long long i = (long long)blockIdx.x * blockDim.x + threadIdx.x;
  for (; i < n; i += (long long)gridDim.x * blockDim.x) p[i] = v;
}

__global__ void deg_scatter_kernel(const int* __restrict__ dst,
                                   const float* __restrict__ w,
                                   float* __restrict__ deg_w,
                                   float* __restrict__ deg_1, int E) {
  int e = blockIdx.x * blockDim.x + threadIdx.x;
  if (e < E) {
    atomicAdd(&deg_w[dst[e]], w[e]);
    atomicAdd(&deg_1[dst[e]], 1.0f);
  }
}

__global__ void dinv_kernel(const float* __restrict__ deg,
                            float* __restrict__ dinv, int n) {
  int i = blockIdx.x * blockDim.x + threadIdx.x;
  if (i < n) {
    float d = deg[i];
    dinv[i] = d > 0.f ? rsqrtf(fmaxf(d, 1e-12f)) : 0.f;
  }
}

// agg[i,f] = dinv[i]^2 * h[i,f]   (self-loop, weight 1)
__global__ void agg_self_kernel(const float* __restrict__ h,
                                const float* __restrict__ dinv,
                                float* __restrict__ agg, int n, int F) {
  long long total = (long long)n * F;
  for (long long idx = (long long)blockIdx.x * blockDim.x + threadIdx.x;
       idx < total; idx += (long long)gridDim.x * blockDim.x) {
    int i = (int)(idx / F);
    float dv = dinv[i];
    agg[idx] = dv * dv * h[idx];
  }
}

// agg[dst,f] += dinv[src]*w*dinv[dst] * h[src,f]
__global__ void agg_edge_kernel(const int* __restrict__ src,
                                const int* __restrict__ dst,
                                const float* __restrict__ w,
                                const float* __restrict__ dinv,
                                const float* __restrict__ h,
                                float* __restrict__ agg, int E, int F) {
  long long total = (long long)E * F;
  for (long long idx = (long long)blockIdx.x * blockDim.x + threadIdx.x;
       idx < total; idx += (long long)gridDim.x * blockDim.x) {
    int e = (int)(idx / F), f = (int)(idx % F);
    int si = src[e], di = dst[e];
    float wt = w ? w[e] : 1.0f;
    float coef = dinv[si] * wt * dinv[di];
    atomicAdd(&agg[(long long)di * F + f], coef * h[(long long)si * F + f]);
  }
}

__global__ void bias_act_kernel(const float* __restrict__ in,
                                const float* __restrict__ b,
                                float* __restrict__ out, int n, int F, int relu) {
  long long total = (long long)n * F;
  for (long long idx = (long long)blockIdx.x * blockDim.x + threadIdx.x;
       idx < total; idx += (long long)gridDim.x * blockDim.x) {
    float v = in[idx] + b[(int)(idx % F)];
    out[idx] = relu ? fmaxf(v, 0.f) : v;
  }
}

// one wave (32 lanes) per row: softmax + entropy accumulation
__global__ void softmax_entropy_kernel(const float* __restrict__ s01,
                                       float* __restrict__ s,
                                       float* __restrict__ ent_sum, int cols) {
  const int row = blockIdx.x;
  const int lane = threadIdx.x;
  const float* in = s01 + (long long)row * cols;
  float* out = s + (long long)row * cols;
  float mx = -3.4e38f;
  for (int c = lane; c < cols; c += 32) mx = fmaxf(mx, in[c]);
  for (int o = 16; o; o >>= 1) mx = fmaxf(mx, __shfl_xor(mx, o, 32));
  float sum = 0.f;
  for (int c = lane; c < cols; c += 32) sum += expf(in[c] - mx);
  for (int o = 16; o; o >>= 1) sum += __shfl_xor(sum, o, 32);
  const float inv = 1.f / sum;
  float ent = 0.f;
  for (int c = lane; c < cols; c += 32) {
    float p = expf(in[c] - mx) * inv;
    out[c] = p;
    ent -= p * logf(p + 1e-15f);
  }
  for (int o = 16; o; o >>= 1) ent += __shfl_xor(ent, o, 32);
  if (lane == 0) atomicAdd(ent_sum, ent);
}

// coarse graph: dinv1 from adj1 with diag fix; then Ahat = dinv1 (x) dinv1 * A1
__global__ void coarse_deg_kernel(const float* __restrict__ adj1,
                                  float* __restrict__ dinv1, int Cn) {
  int c = blockIdx.x * blockDim.x + threadIdx.x;
  if (c < Cn) {
    float diag = adj1[(long long)c * Cn + c];
    float dfix = (diag == 0.f) ? 1.f : diag;
    float sum = 0.f;
    for (int d = 0; d < Cn; ++d)
      sum += (d == c) ? dfix : adj1[(long long)c * Cn + d];
    dinv1[c] = sum > 0.f ? rsqrtf(fmaxf(sum, 1e-12f)) : 0.f;
  }
}

__global__ void ahat_kernel(const float* __restrict__ adj1,
                            const float* __restrict__ dinv1,
                            float* __restrict__ Ahat, int Cn) {
  int idx = blockIdx.x * blockDim.x + threadIdx.x;
  if (idx < Cn * Cn) {
    int c = idx / Cn, d = idx % Cn;
    float v = adj1[idx];
    if (c == d && v == 0.f) v = 1.f;
    Ahat[idx] = dinv1[c] * v * dinv1[d];
  }
}

__global__ void logsoftmax_kernel(const float* __restrict__ in,
                                  float* __restrict__ out, int n, int cols) {
  int row = blockIdx.x * blockDim.x + threadIdx.x;
  if (row >= n) return;
  const float* r = in + (long long)row * cols;
  float mx = r[0];
  for (int c = 1; c < cols; ++c) mx = fmaxf(mx, r[c]);
  float s = 0.f;
  for (int c = 0; c < cols; ++c) s += expf(r[c] - mx);
  float ls = logf(s);
  float* o = out + (long long)row * cols;
  for (int c = 0; c < cols; ++c) o[c] = r[c] - mx - ls;
}

// scalar = sqrt(sum_adj2 - 2*trace(adj1) + ||G||_F^2)/N^2 + ent_sum/N
__global__ void finalize_kernel(const float* __restrict__ G,
                                const float* __restrict__ adj1,
                                const float* __restrict__ scal,
                                float* __restrict__ out_scalar) {
  __shared__ float red[256];
  const int t = threadIdx.x;
  float g2 = 0.f;
  for (int i = t; i < C_ * C_; i += 256) { float v = G[i]; g2 += v * v; }
  red[t] = g2; __syncthreads();
  for (int o = 128; o; o >>= 1) { if (t < o) red[t] += red[t + o]; __syncthreads(); }
  const float sumG2 = red[0]; __syncthreads();
  float tr = 0.f;
  for (int c = t; c < C_; c += 256) tr += adj1[c * C_ + c];
  red[t] = tr; __syncthreads();
  for (int o = 128; o; o >>= 1) { if (t < o) red[t] += red[t + o]; __syncthreads(); }
  const float trace = red[0];
  if (t == 0) {
    float link2 = scal[0] - 2.f * trace + sumG2;
    float l1 = sqrtf(fmaxf(link2, 0.f)) / ((float)N_ * (float)N_);
    float e1 = scal[1] / (float)N_;
    out_scalar[0] = l1 + e1;
  }
}

// ---------------------------------------------------------------------------
extern "C" void kernel_launch(void* const* d_in, const int* in_sizes, int n_in,
                              void* d_out, int out_size, void* d_ws,
                              size_t ws_size, hipStream_t stream) {
  (void)in_sizes; (void)n_in; (void)out_size; (void)ws_size;
  const float* x   = (const float*)d_in[0];
  const int*   ei  = (const int*)d_in[1];      // [2,E] int32 (JAX x64 off)
  const float* ew  = (const float*)d_in[2];
  const float* adj = (const float*)d_in[3];
  const float* W0a = (const float*)d_in[5];
  const float* b0a = (const float*)d_in[6];
  const float* W0b = (const float*)d_in[7];
  const float* b0b = (const float*)d_in[8];
  const float* Wp  = (const float*)d_in[9];
  const float* bp  = (const float*)d_in[10];
  const float* W1a = (const float*)d_in[11];
  const float* b1a = (const float*)d_in[12];
  const float* W1b = (const float*)d_in[13];
  const float* b1b = (const float*)d_in[14];
  const float* Wc  = (const float*)d_in[15];
  const float* bc  = (const float*)d_in[16];
  const int* src = ei;
  const int* dst = ei + E_;

  float* out      = (float*)d_out;
  float* pred_out = out;                                   // [N,10]
  float* s01_out  = out + (long long)N_ * NCLS_;           // [N,100]
  float* scal_out = s01_out + (long long)N_ * C_;          // [1]
  float* adj1_out = scal_out + 1;                          // [100,100]

  float* ws = (float*)d_ws;
  float* w_deg_w = ws;                float* w_dinv_w = w_deg_w + N_;
  float* w_deg_1 = w_dinv_w + N_;     float* w_dinv_1 = w_deg_1 + N_;
  float* w_tmp16 = w_dinv_1 + N_;                      // N*16
  float* w_agg16 = w_tmp16 + (long long)N_ * NH_;
  float* w_h0    = w_agg16 + (long long)N_ * NH_;
  float* w_hW0b  = w_h0 + (long long)N_ * NH_;         // N*10
  float* w_agg10 = w_hW0b + (long long)N_ * NO0_;
  float* w_x0    = w_agg10 + (long long)N_ * NO0_;
  float* w_xWp   = w_x0 + (long long)N_ * NO0_;        // N*100
  float* w_aggP  = w_xWp + (long long)N_ * C_;
  float* w_s     = w_aggP + (long long)N_ * C_;
  float* w_x1p   = w_s + (long long)N_ * C_;           // 100*512
  float* w_G     = w_x1p + C_ * F_IN;
  float* w_Ahat  = w_G + C_ * C_;
  float* w_dinv1 = w_Ahat + C_ * C_;                   // 128
  float* w_xW1a  = w_dinv1 + 128;
  float* w_h1    = w_xW1a + C_ * NH_;
  float* w_xW1b  = w_h1 + C_ * NH_;
  float* w_x1    = w_xW1b + C_ * C_;
  float* w_hWc   = w_x1 + C_ * C_;                     // N*10
  float* w_aggF  = w_hWc + (long long)N_ * NCLS_;
  float* w_outF  = w_aggF + (long long)N_ * NCLS_;
  float* w_scal  = w_outF + (long long)N_ * NCLS_;     // [0]=sum_adj2 [1]=ent
  float* w_sTp   = w_scal + 16;                        // [112][8192] padded s^T
  float* w_t     = w_aggP;   // alias: free after s01 is produced
  float* w_x1u   = w_xWp;    // alias: free after s01 is produced

  auto blocks = [](long long n) {
    long long b = (n + 255) / 256;
    return (int)(b > 32768 ? 32768 : b);
  };

  // ---- degrees (with self loops) -----------------------------------------
  fill_kernel<<<blocks(N_), 256, 0, stream>>>(w_deg_w, 1.0f, N_);
  fill_kernel<<<blocks(N_), 256, 0, stream>>>(w_deg_1, 1.0f, N_);
  fill_kernel<<<1, 256, 0, stream>>>(w_scal, 0.0f, 8);
  deg_scatter_kernel<<<(E_ + 255) / 256, 256, 0, stream>>>(dst, ew, w_deg_w,
                                                           w_deg_1, E_);
  dinv_kernel<<<(N_ + 255) / 256, 256, 0, stream>>>(w_deg_w, w_dinv_w, N_);
  dinv_kernel<<<(N_ + 255) / 256, 256, 0, stream>>>(w_deg_1, w_dinv_1, N_);

  // ---- gcn0 layer A: h0 = relu(agg(x@W0a) + b0a) -------------------------
  fill_kernel<<<blocks((long long)N_ * NH_), 256, 0, stream>>>(
      w_tmp16, 0.f, (long long)N_ * NH_);
  run_gemm<1, false, false>(x, W0a, nullptr, w_tmp16, N_, NH_, F_IN, F_IN, NH_,
                            NH_, 4, stream);
  agg_self_kernel<<<blocks((long long)N_ * NH_), 256, 0, stream>>>(
      w_tmp16, w_dinv_w, w_agg16, N_, NH_);
  agg_edge_kernel<<<blocks((long long)E_ * NH_), 256, 0, stream>>>(
      src, dst, ew, w_dinv_w, w_tmp16, w_agg16, E_, NH_);
  bias_act_kernel<<<blocks((long long)N_ * NH_), 256, 0, stream>>>(
      w_agg16, b0a, w_h0, N_, NH_, 1);

  // ---- gcn0 layer B: x0 = agg(h0@W0b) + b0b ------------------------------
  run_gemm<0, false, false>(w_h0, W0b, nullptr, w_hW0b, N_, NO0_, NH_, NH_,
                            NO0_, NO0_, 1, stream);
  agg_self_kernel<<<blocks((long long)N_ * NO0_), 256, 0, stream>>>(
      w_hW0b, w_dinv_w, w_agg10, N_, NO0_);
  agg_edge_kernel<<<blocks((long long)E_ * NO0_), 256, 0, stream>>>(
      src, dst, ew, w_dinv_w, w_hW0b, w_agg10, E_, NO0_);
  bias_act_kernel<<<blocks((long long)N_ * NO0_), 256, 0, stream>>>(
      w_agg10, b0b, w_x0, N_, NO0_, 0);

  // ---- s01 = relu(agg(x@Wp) + bp) ----------------------------------------
  run_gemm<0, false, false>(x, Wp, nullptr, w_xWp, N_, C_, F_IN, F_IN, C_, C_,
                            1, stream);
  agg_self_kernel<<<blocks((long long)N_ * C_), 256, 0, stream>>>(
      w_xWp, w_dinv_w, w_aggP, N_, C_);
  agg_edge_kernel<<<blocks((long long)E_ * C_), 256, 0, stream>>>(
      src, dst, ew, w_dinv_w, w_xWp, w_aggP, E_, C_);
  bias_act_kernel<<<blocks((long long)N_ * C_), 256, 0, stream>>>(
      w_aggP, bp, s01_out, N_, C_, 1);

  // ---- softmax s + entropy; padded transpose of s ------------------------
  softmax_entropy_kernel<<<N_, 32, 0, stream>>>(s01_out, w_s, w_scal + 1, C_);
  transpose_pad_kernel<<<blocks((long long)CPAD_ * N_), 256, 0, stream>>>(
      w_s, w_sTp);

  // ---- t = adj @ s (fused sum(adj^2)); adj streamed once -----------------
  fill_kernel<<<blocks((long long)N_ * C_), 256, 0, stream>>>(
      w_t, 0.f, (long long)N_ * C_);
  {
    const int kch = 4;                       // 2048 waves total
    const int waves = (N_ / 16) * kch;
    adj_s_wmma<<<waves / 4, 128, 0, stream>>>(adj, w_sTp, w_t, w_scal,
                                              N_ / kch);
  }

  // ---- adj1 = s'@t ; x1p = s'@x ; G = s'@s  (A = sTp, row-major) ---------
  fill_kernel<<<blocks(C_ * C_), 256, 0, stream>>>(adj1_out, 0.f, C_ * C_);
  run_gemm<1, false, false>(w_sTp, w_t, nullptr, adj1_out, C_, C_, N_, N_, C_,
                            C_, 32, stream);
  fill_kernel<<<blocks(C_ * F_IN), 256, 0, stream>>>(w_x1p, 0.f, C_ * F_IN);
  run_gemm<1, false, false>(w_sTp, x, nullptr, w_x1p, C_, F_IN, N_, N_, F_IN,
                            F_IN, 8, stream);
  fill_kernel<<<blocks(C_ * C_), 256, 0, stream>>>(w_G, 0.f, C_ * C_);
  run_gemm<1, false, false>(w_sTp, w_s, nullptr, w_G, C_, C_, N_, N_, C_, C_,
                            32, stream);

  // ---- coarse dense GCN ---------------------------------------------------
  coarse_deg_kernel<<<1, 128, 0, stream>>>(adj1_out, w_dinv1, C_);
  ahat_kernel<<<(C_ * C_ + 255) / 256, 256, 0, stream>>>(adj1_out, w_dinv1,
                                                         w_Ahat, C_);
  run_gemm<0, false, false>(w_x1p, W1a, nullptr, w_xW1a, C_, NH_, F_IN, F_IN,
                            NH_, NH_, 1, stream);
  run_gemm<0, true, true>(w_Ahat, w_xW1a, b1a, w_h1, C_, NH_, C_, C_, NH_, NH_,
                          1, stream);
  run_gemm<0, false, false>(w_h1, W1b, nullptr, w_xW1b, C_, C_, NH_, NH_, C_,
                            C_, 1, stream);
  run_gemm<0, true, false>(w_Ahat, w_xW1b, b1b, w_x1, C_, C_, C_, C_, C_, C_,
                           1, stream);

  // ---- unpool + final features: hWc = x0@Wc[0:10] + (s01@x1)@Wc[10:110] --
  run_gemm<0, false, false>(s01_out, w_x1, nullptr, w_x1u, N_, C_, C_, C_, C_,
                            C_, 1, stream);
  run_gemm<0, false, false>(w_x0, Wc, nullptr, w_hWc, N_, NCLS_, NO0_, NO0_,
                            NCLS_, NCLS_, 1, stream);
  run_gemm<2, false, false>(w_x1u, Wc + NO0_ * NCLS_, nullptr, w_hWc, N_,
                            NCLS_, C_, C_, NCLS_, NCLS_, 1, stream);

  // ---- final sparse GCN (edge weights = 1) + log_softmax -----------------
  agg_self_kernel<<<blocks((long long)N_ * NCLS_), 256, 0, stream>>>(
      w_hWc, w_dinv_1, w_aggF, N_, NCLS_);
  agg_edge_kernel<<<blocks((long long)E_ * NCLS_), 256, 0, stream>>>(
      src, dst, nullptr, w_dinv_1, w_hWc, w_aggF, E_, NCLS_);
  bias_act_kernel<<<blocks((long long)N_ * NCLS_), 256, 0, stream>>>(
      w_aggF, bc, w_outF, N_, NCLS_, 0);
  logsoftmax_kernel<<<(N_ + 255) / 256, 256, 0, stream>>>(w_outF, pred_out, N_,
                                                          NCLS_);

  // ---- scalar: l1 + e1 ----------------------------------------------------
  finalize_kernel<<<1, 256, 0, stream>>>(w_G, adj1_out, w_scal, scal_out);
}